// GAT3_26465588478556
// MI455X (gfx1250) — compile-verified
//
#include <hip/hip_runtime.h>
#include <hip/hip_bf16.h>
#include <math.h>

typedef __attribute__((ext_vector_type(2))) float v2f;
typedef __attribute__((ext_vector_type(8))) float v8f;

#define GAT_SLOPE 0.2f
#define BN_EPS    1e-5f

static __device__ __forceinline__ void atomicMaxF(float* a, float v) {
  // monotonic int-punning float max (works for mixed signs, init = -inf)
  if (v >= 0.f) atomicMax((int*)a, __float_as_int(v));
  else          atomicMin((unsigned int*)a, __float_as_uint(v));
}

__global__ void k_fill(float* __restrict__ p, float v, int n) {
  int i = blockIdx.x * blockDim.x + threadIdx.x;
  if (i < n) p[i] = v;
}

__global__ void k_bias_init(float* __restrict__ h, const float* __restrict__ bias, int n64) {
  int i = blockIdx.x * blockDim.x + threadIdx.x;
  if (i < n64) h[i] = bias[i & 63];
}

// self-loop attr: cnt[d] += 1, asum[d] += ea[e]
__global__ void k_selfloop_stats(const int* __restrict__ dst, const float* __restrict__ ea,
                                 float* __restrict__ cnt, float* __restrict__ asum, int E) {
  int e = blockIdx.x * blockDim.x + threadIdx.x;
  if (e < E) {
    int d = dst[e];
    atomicAdd(&cnt[d], 1.f);
    atomicAdd(&asum[d], ea[e]);
  }
}

__global__ void k_mean_attr(const float* __restrict__ cnt, const float* __restrict__ asum,
                            float* __restrict__ mattr, int N) {
  int i = blockIdx.x * blockDim.x + threadIdx.x;
  if (i < N) mattr[i] = asum[i] / fmaxf(cnt[i], 1.f);
}

// out[N,64] = X[N,K] @ W[K,64] + bias (+ accb) ; act: 0=none 1=relu
// One wave per 16x16 C tile, fp32 WMMA (16x16x4), K in {32,64}.
__global__ void k_gemm_wmma(const float* __restrict__ X, const float* __restrict__ W,
                            const float* __restrict__ bias, const float* __restrict__ accb,
                            float* __restrict__ out, int N, int K, int act) {
  const int wave = threadIdx.x >> 5;
  const int lane = threadIdx.x & 31;
  const int half = lane >> 4;       // which 16-lane half
  const int lid  = lane & 15;
  const int m0 = blockIdx.x * 16;
  const int n0 = wave * 16;         // 4 waves -> 4 column tiles of 64-wide C
  int arow = m0 + lid; if (arow >= N) arow = N - 1;   // clamp (no divergence before wmma)
  const float* __restrict__ xrow = X + (size_t)arow * K;
  const int col = n0 + lid;
  v8f acc = {0.f,0.f,0.f,0.f,0.f,0.f,0.f,0.f};
  for (int k = 0; k < K; k += 4) {
    const int kk = k + half * 2;    // half 0 -> K={k,k+1}, half 1 -> K={k+2,k+3}
    v2f a; a.x = xrow[kk];            a.y = xrow[kk + 1];
    v2f b; b.x = W[kk * 64 + col];    b.y = W[(kk + 1) * 64 + col];
    acc = __builtin_amdgcn_wmma_f32_16x16x4_f32(false, a, false, b, (short)0, acc, false, false);
  }
  const float bv = bias[col];
  #pragma unroll
  for (int r = 0; r < 8; ++r) {
    const int row = m0 + half * 8 + r;   // ISA C/D layout: VGPR r, halves hold M=r / M=8+r
    if (row < N) {
      float v = acc[r] + bv;
      if (accb) v += accb[(size_t)row * 64 + col];
      if (act == 1) v = fmaxf(v, 0.f);
      out[(size_t)row * 64 + col] = v;
    }
  }
}

// one wave per edge; lane covers channels c and c+32
__global__ void k_edge_logit(const int* __restrict__ src, const int* __restrict__ dst,
                             const float* __restrict__ ea, const float* __restrict__ mattr,
                             const float* __restrict__ xl, const float* __restrict__ xr,
                             const float* __restrict__ We, const float* __restrict__ att,
                             float* __restrict__ logit, float* __restrict__ mx, int E, int EP) {
  int e = blockIdx.x * (blockDim.x >> 5) + (threadIdx.x >> 5);
  if (e >= EP) return;
  int lane = threadIdx.x & 31;
  int s, d; float a;
  if (e < E) { s = src[e]; d = dst[e]; a = ea[e]; }
  else       { s = e - E;  d = s;      a = mattr[s]; }
  float p = 0.f;
  #pragma unroll
  for (int j = 0; j < 2; ++j) {
    int c = lane + j * 32;
    float v = xl[(size_t)s * 64 + c] + xr[(size_t)d * 64 + c] + a * We[c];
    v = (v > 0.f) ? v : GAT_SLOPE * v;          // leaky_relu(0.2)
    p += v * att[c];
  }
  #pragma unroll
  for (int ofs = 16; ofs; ofs >>= 1) p += __shfl_down(p, ofs, 32);
  if (lane == 0) { logit[e] = p; atomicMaxF(&mx[d], p); }
}

__global__ void k_edge_exp(const int* __restrict__ dst, const float* __restrict__ logit,
                           const float* __restrict__ mx, float* __restrict__ exv,
                           float* __restrict__ den, int E, int EP) {
  int e = blockIdx.x * blockDim.x + threadIdx.x;
  if (e >= EP) return;
  int d = (e < E) ? dst[e] : e - E;
  float v = expf(logit[e] - mx[d]);
  exv[e] = v;
  atomicAdd(&den[d], v);
}

__global__ void k_edge_aggregate(const int* __restrict__ src, const int* __restrict__ dst,
                                 const float* __restrict__ exv, const float* __restrict__ den,
                                 const float* __restrict__ xl, float* __restrict__ h,
                                 float* __restrict__ alpha_out, int E, int EP) {
  int e = blockIdx.x * (blockDim.x >> 5) + (threadIdx.x >> 5);
  if (e >= EP) return;
  int lane = threadIdx.x & 31;
  int s, d;
  if (e < E) { s = src[e]; d = dst[e]; } else { s = e - E; d = s; }
  float alpha = exv[e] / den[d];
  if (alpha_out && lane == 0) alpha_out[e] = alpha;
  #pragma unroll
  for (int j = 0; j < 2; ++j) {
    int c = lane + j * 32;
    atomicAdd(&h[(size_t)d * 64 + c], alpha * xl[(size_t)s * 64 + c]);
  }
}

__global__ void k_bn_stats(const float* __restrict__ h, float* __restrict__ stats, int N) {
  int gid = blockIdx.x * blockDim.x + threadIdx.x;
  int c = gid & 63;
  int r = gid >> 6;
  int stride = (gridDim.x * blockDim.x) >> 6;
  float s = 0.f, s2 = 0.f;
  for (; r < N; r += stride) {
    float v = h[(size_t)r * 64 + c];
    s += v; s2 += v * v;
  }
  atomicAdd(&stats[c], s);
  atomicAdd(&stats[64 + c], s2);
}

__global__ void k_bn_apply_elu(const float* __restrict__ in, const float* __restrict__ stats,
                               const float* __restrict__ gamma, const float* __restrict__ beta,
                               float* __restrict__ outp, int N) {
  int i = blockIdx.x * blockDim.x + threadIdx.x;
  if (i >= N * 64) return;
  int c = i & 63;
  float invN = 1.f / (float)N;
  float mu  = stats[c] * invN;
  float var = stats[64 + c] * invN - mu * mu;   // biased var
  float y = (in[i] - mu) * rsqrtf(var + BN_EPS) * gamma[c] + beta[c];
  outp[i] = (y > 0.f) ? y : (expf(y) - 1.f);    // elu
}

// classifier heads + fused log_softmax; weights staged in LDS
__global__ void k_heads(const float* __restrict__ hl, const float* __restrict__ Wc,
                        const float* __restrict__ bc, const float* __restrict__ Wt,
                        const float* __restrict__ bt, float* __restrict__ out, int N) {
  __shared__ float sWc[640], sbc[10], sWt[1280], sbt[20];
  for (int i = threadIdx.x; i < 640; i += blockDim.x)  sWc[i] = Wc[i];
  for (int i = threadIdx.x; i < 1280; i += blockDim.x) sWt[i] = Wt[i];
  if (threadIdx.x < 10) sbc[threadIdx.x] = bc[threadIdx.x];
  if (threadIdx.x < 20) sbt[threadIdx.x] = bt[threadIdx.x];
  __syncthreads();
  int i = blockIdx.x * blockDim.x + threadIdx.x;
  if (i >= N) return;
  float hr[64];
  #pragma unroll
  for (int k = 0; k < 64; ++k) hr[k] = hl[(size_t)i * 64 + k];
  float* orow = out + (size_t)i * 30;

  float lc[10];
  #pragma unroll
  for (int j = 0; j < 10; ++j) {
    float s = sbc[j];
    for (int k = 0; k < 64; ++k) s += hr[k] * sWc[k * 10 + j];
    lc[j] = s;
  }
  float m = lc[0];
  #pragma unroll
  for (int j = 1; j < 10; ++j) m = fmaxf(m, lc[j]);
  float ssum = 0.f;
  #pragma unroll
  for (int j = 0; j < 10; ++j) ssum += expf(lc[j] - m);
  float lg = logf(ssum);
  #pragma unroll
  for (int j = 0; j < 10; ++j) orow[j] = lc[j] - m - lg;

  float lt[20];
  #pragma unroll
  for (int j = 0; j < 20; ++j) {
    float s = sbt[j];
    for (int k = 0; k < 64; ++k) s += hr[k] * sWt[k * 20 + j];
    lt[j] = s;
  }
  m = lt[0];
  #pragma unroll
  for (int j = 1; j < 20; ++j) m = fmaxf(m, lt[j]);
  ssum = 0.f;
  #pragma unroll
  for (int j = 0; j < 20; ++j) ssum += expf(lt[j] - m);
  lg = logf(ssum);
  #pragma unroll
  for (int j = 0; j < 20; ++j) orow[10 + j] = lt[j] - m - lg;
}

extern "C" void kernel_launch(void* const* d_in, const int* in_sizes, int n_in,
                              void* d_out, int out_size, void* d_ws, size_t ws_size,
                              hipStream_t stream) {
  const float* x    = (const float*)d_in[0];
  const int*   ei   = (const int*)  d_in[1];
  const float* ea   = (const float*)d_in[2];
  const float* Wl0  = (const float*)d_in[3];
  const float* bl0  = (const float*)d_in[4];
  const float* Wr0  = (const float*)d_in[5];
  const float* br0  = (const float*)d_in[6];
  const float* We0  = (const float*)d_in[7];
  const float* att0 = (const float*)d_in[8];
  const float* bg0  = (const float*)d_in[9];
  const float* Wl1  = (const float*)d_in[10];
  const float* bl1  = (const float*)d_in[11];
  const float* Wr1  = (const float*)d_in[12];
  const float* br1  = (const float*)d_in[13];
  const float* We1  = (const float*)d_in[14];
  const float* att1 = (const float*)d_in[15];
  const float* bg1  = (const float*)d_in[16];
  const float* Ws0  = (const float*)d_in[17];
  const float* bs0  = (const float*)d_in[18];
  const float* Ws1  = (const float*)d_in[19];
  const float* bs1  = (const float*)d_in[20];
  const float* g0   = (const float*)d_in[21];
  const float* be0  = (const float*)d_in[22];
  const float* g1   = (const float*)d_in[23];
  const float* be1  = (const float*)d_in[24];
  const float* Wf   = (const float*)d_in[25];
  const float* bf   = (const float*)d_in[26];
  const float* Wc   = (const float*)d_in[27];
  const float* bc   = (const float*)d_in[28];
  const float* Wt   = (const float*)d_in[29];
  const float* bt   = (const float*)d_in[30];
  (void)n_in; (void)out_size; (void)ws_size;

  const int N  = in_sizes[0] / 32;
  const int E  = in_sizes[1] / 2;
  const int EP = E + N;
  const int* src = ei;        // edge_index row 0
  const int* dst = ei + E;    // edge_index row 1

  float* ws = (float*)d_ws;
  size_t o = 0;
  float* xl    = ws + o; o += (size_t)N * 64;
  float* xr    = ws + o; o += (size_t)N * 64;
  float* h     = ws + o; o += (size_t)N * 64;
  float* h2    = ws + o; o += (size_t)N * 64;
  float* logit = ws + o; o += (size_t)EP;
  float* exv   = ws + o; o += (size_t)EP;
  float* mx    = ws + o; o += (size_t)N;
  float* den   = ws + o; o += (size_t)N;
  float* cnt   = ws + o; o += (size_t)N;
  float* asum  = ws + o; o += (size_t)N;
  float* mattr = ws + o; o += (size_t)N;
  float* stats = ws + o; o += 128;

  float* out       = (float*)d_out;
  float* alpha_out = out + (size_t)N * 30;   // second return value w = alpha (layer 2)

  const int T = 256;
  auto cdiv = [](int a, int b) { return (a + b - 1) / b; };
  const int gN    = cdiv(N, T);
  const int gN64  = cdiv(N * 64, T);
  const int gEP   = cdiv(EP, T);
  const int gWave = cdiv(EP, T / 32);        // one wave per edge
  dim3 gemmGrid(cdiv(N, 16)), gemmBlk(128);

  // self-loop mean edge attr (fill_value='mean')
  k_fill<<<gN, T, 0, stream>>>(cnt, 0.f, N);
  k_fill<<<gN, T, 0, stream>>>(asum, 0.f, N);
  k_selfloop_stats<<<cdiv(E, T), T, 0, stream>>>(dst, ea, cnt, asum, E);
  k_mean_attr<<<gN, T, 0, stream>>>(cnt, asum, mattr, N);

  // ---------------- layer 1 (input x, K=32) ----------------
  k_gemm_wmma<<<gemmGrid, gemmBlk, 0, stream>>>(x, Wl0, bl0, nullptr, xl, N, 32, 0);
  k_gemm_wmma<<<gemmGrid, gemmBlk, 0, stream>>>(x, Wr0, br0, nullptr, xr, N, 32, 0);
  k_fill<<<gN, T, 0, stream>>>(mx, -INFINITY, N);
  k_fill<<<gN, T, 0, stream>>>(den, 0.f, N);
  k_edge_logit<<<gWave, T, 0, stream>>>(src, dst, ea, mattr, xl, xr, We0, att0, logit, mx, E, EP);
  k_edge_exp<<<gEP, T, 0, stream>>>(dst, logit, mx, exv, den, E, EP);
  k_bias_init<<<gN64, T, 0, stream>>>(h, bg0, N * 64);
  k_edge_aggregate<<<gWave, T, 0, stream>>>(src, dst, exv, den, xl, h, nullptr, E, EP);
  // residual skip: h = h + x@Ws0 + bs0
  k_gemm_wmma<<<gemmGrid, gemmBlk, 0, stream>>>(x, Ws0, bs0, h, h, N, 32, 0);
  k_fill<<<1, 128, 0, stream>>>(stats, 0.f, 128);
  k_bn_stats<<<1024, 256, 0, stream>>>(h, stats, N);
  k_bn_apply_elu<<<gN64, T, 0, stream>>>(h, stats, g0, be0, h2, N);

  // ---------------- layer 2 (input h2, K=64) ----------------
  k_gemm_wmma<<<gemmGrid, gemmBlk, 0, stream>>>(h2, Wl1, bl1, nullptr, xl, N, 64, 0);
  k_gemm_wmma<<<gemmGrid, gemmBlk, 0, stream>>>(h2, Wr1, br1, nullptr, xr, N, 64, 0);
  k_fill<<<gN, T, 0, stream>>>(mx, -INFINITY, N);
  k_fill<<<gN, T, 0, stream>>>(den, 0.f, N);
  k_edge_logit<<<gWave, T, 0, stream>>>(src, dst, ea, mattr, xl, xr, We1, att1, logit, mx, E, EP);
  k_edge_exp<<<gEP, T, 0, stream>>>(dst, logit, mx, exv, den, E, EP);
  k_bias_init<<<gN64, T, 0, stream>>>(h, bg1, N * 64);
  k_edge_aggregate<<<gWave, T, 0, stream>>>(src, dst, exv, den, xl, h, alpha_out, E, EP);
  // non-residual skip on NEW h (as in reference): xl = h + h@Ws1 + bs1  (xl is free now)
  k_gemm_wmma<<<gemmGrid, gemmBlk, 0, stream>>>(h, Ws1, bs1, h, xl, N, 64, 0);
  k_fill<<<1, 128, 0, stream>>>(stats, 0.f, 128);
  k_bn_stats<<<1024, 256, 0, stream>>>(xl, stats, N);
  k_bn_apply_elu<<<gN64, T, 0, stream>>>(xl, stats, g1, be1, h2, N);

  // fc1 + relu: h = relu(h2 @ Wf + bf)
  k_gemm_wmma<<<gemmGrid, gemmBlk, 0, stream>>>(h2, Wf, bf, nullptr, h, N, 64, 1);
  // classifier heads + log_softmax -> out[:, 0:30]
  k_heads<<<cdiv(N, 128), 128, 0, stream>>>(h, Wc, bc, Wt, bt, out, N);
}